// Face_atten_14310831030500
// MI455X (gfx1250) — compile-verified
//
#include <hip/hip_runtime.h>
#include <math.h>
#include <stdint.h>

#define D_MODEL 256
#define THREE_D 768
#define NHEAD   2
#define HDIM    128
#define LTOT    8192
#define BSAMP   64
#define FFACE   128
#define EPSV    1e-5f

// padded LDS strides (floats): stride 132 -> bank = 4*idx + k, conflict-free
// for the NT fragment pattern (lanes 0-15 read k, lanes 16-31 read k+2).
#define KLD 132
#define VTL 132
#define SLD 132

// LDS byte layout of the attention kernel (dynamic smem starts at offset 0)
#define K_LDS_OFF   0
#define VT_LDS_OFF  (128 * KLD * 4)
#define S_LDS_OFF   (VT_LDS_OFF + 128 * VTL * 4)
#define ATTN_LDS_BYTES (S_LDS_OFF + 128 * SLD * 4)

// fused out-proj kernel LDS tile: 64 rows x 256 cols, stride 260 floats
#define TLD 260
#define OUTLN_LDS_BYTES (64 * TLD * 4)

typedef __attribute__((ext_vector_type(2))) float v2f;
typedef __attribute__((ext_vector_type(8))) float v8f;
typedef __attribute__((ext_vector_type(4))) unsigned int u32x4;
typedef __attribute__((ext_vector_type(4))) int i32x4;
typedef __attribute__((ext_vector_type(8))) int i32x8;

__device__ __forceinline__ v8f wmma_f32(v2f a, v2f b, v8f c) {
    // D = A(16x4 f32) * B(4x16 f32) + C(16x16 f32), full fp32 precision
    return __builtin_amdgcn_wmma_f32_16x16x4_f32(
        false, a, false, b, (short)0, c, false, false);
}

// ---------------------------------------------------------------------------
// Prefix-sum of per-sample face counts from the packed boolean mask.
// ---------------------------------------------------------------------------
__global__ void prep_kernel(const unsigned char* __restrict__ mask,
                            int* __restrict__ starts, int* __restrict__ counts) {
    int t = threadIdx.x;
    if (t < BSAMP) {
        int c = 0;
        for (int f = 0; f < FFACE; ++f) c += (mask[t * FFACE + f] != 0);
        counts[t] = c;
    }
    __syncthreads();
    if (t == 0) {
        int run = 0;
        for (int b = 0; b < BSAMP; ++b) { starts[b] = run; run += counts[b]; }
    }
}

// ---------------------------------------------------------------------------
// NT GEMM: C[M,N] = A[M,K] * W[N,K]^T + bias[N]   (QKV projection)
// grid = (M/128, N/128), block = 256 (8 waves as 4x2).
// Each wave: 32x64 strip = 2x4 tiles of 16x16; software-pipelined K loop.
// ---------------------------------------------------------------------------
template <int K>
__global__ __launch_bounds__(256) void gemm_nt_bias(
    const float* __restrict__ A, int lda,
    const float* __restrict__ W,      // [N, K] row-major
    const float* __restrict__ bias,   // [N]
    float* __restrict__ C, int ldc)
{
    const int wave = threadIdx.x >> 5;
    const int lane = threadIdx.x & 31;
    const int ml   = lane & 15;
    const int kh   = lane >> 4;
    const int wm   = wave & 3, wn = wave >> 2;
    const int m0   = blockIdx.x * 128 + wm * 32;
    const int n0   = blockIdx.y * 128 + wn * 64;

    const float* a0 = A + (long)(m0 + ml) * lda + 2 * kh;
    const float* a1 = a0 + 16 * lda;
    const float* w0 = W + (long)(n0 + ml) * K + 2 * kh;

    v8f acc[2][4] = {};
    v2f ac0 = *(const v2f*)(a0);
    v2f ac1 = *(const v2f*)(a1);
    v2f bc[4];
#pragma unroll
    for (int t = 0; t < 4; ++t) bc[t] = *(const v2f*)(w0 + t * 16 * K);

    for (int k = 4; k < K; k += 4) {          // prefetch k, compute k-4
        v2f an0 = *(const v2f*)(a0 + k);
        v2f an1 = *(const v2f*)(a1 + k);
        v2f bn[4];
#pragma unroll
        for (int t = 0; t < 4; ++t) bn[t] = *(const v2f*)(w0 + t * 16 * K + k);
#pragma unroll
        for (int t = 0; t < 4; ++t) {
            acc[0][t] = wmma_f32(ac0, bc[t], acc[0][t]);
            acc[1][t] = wmma_f32(ac1, bc[t], acc[1][t]);
        }
        ac0 = an0; ac1 = an1;
#pragma unroll
        for (int t = 0; t < 4; ++t) bc[t] = bn[t];
    }
#pragma unroll
    for (int t = 0; t < 4; ++t) {
        acc[0][t] = wmma_f32(ac0, bc[t], acc[0][t]);
        acc[1][t] = wmma_f32(ac1, bc[t], acc[1][t]);
    }

#pragma unroll
    for (int t = 0; t < 4; ++t) {
        int col = n0 + t * 16 + ml;
        float bv = bias[col];
#pragma unroll
        for (int rm = 0; rm < 2; ++rm)
#pragma unroll
            for (int r = 0; r < 8; ++r)
                C[(long)(m0 + rm * 16 + r + 8 * kh) * ldc + col] = acc[rm][t][r] + bv;
    }
}

// ---------------------------------------------------------------------------
// Block-diagonal attention: one workgroup per (sample, head).
// K tile DMA'd into LDS by the Tensor Data Mover (padded stride + OOB zero
// fill); V staged transposed; scores in LDS for softmax + second GEMM.
// ---------------------------------------------------------------------------
__global__ __launch_bounds__(256) void attn_kernel(
    const float* __restrict__ qkv,    // [L, 768] : [q | k | v], head h at +h*128
    float* __restrict__ O,            // [L, 256]
    const int* __restrict__ starts, const int* __restrict__ counts)
{
    extern __shared__ float smem[];
    float* Kl = smem;                                  // [128][KLD]  @ byte 0
    float* Vt = smem + 128 * KLD;                      // [128][VTL]  (V transposed)
    float* S  = smem + 128 * KLD + 128 * VTL;          // [128][SLD]

    const int b    = blockIdx.x;
    const int h    = blockIdx.y;
    const int row0 = starts[b];
    const int len  = counts[b];
    const int tid  = threadIdx.x;

#if __has_builtin(__builtin_amdgcn_tensor_load_to_lds)
    // ---- TDM: DMA K tile [128 x 128 f32] global -> LDS, row pad 4 DWORDs
    // (LDS stride 132 floats), rows >= len zero-filled via tensor_dim1 = len.
    if (tid < 32) {   // wave 0 only: TENSORcnt is per-wave
        uint64_t gaddr = (uint64_t)(uintptr_t)(qkv + (long)row0 * THREE_D
                                               + D_MODEL + h * HDIM);
        u32x4 g0;
        g0.x = 1u;                                   // count = 1 valid descriptor
        g0.y = (unsigned)K_LDS_OFF;                  // lds_addr (bytes)
        g0.z = (unsigned)(gaddr & 0xFFFFFFFFu);      // global_addr[31:0]
        g0.w = (unsigned)((gaddr >> 32) & 0x01FFFFFFu) | 0x80000000u; // addr[56:32] | type=2

        i32x8 g1;
        g1[0] = (int)0x07920000u;   // data_size=4B, pad_enable, interval=128dw, amount=4dw
        g1[1] = (int)(128u << 16);  // tensor_dim0 = 128
        g1[2] = (int)(((unsigned)len & 0xFFFFu) << 16);  // tensor_dim1 = len (OOB rows -> 0)
        g1[3] = (int)(128u << 16);  // tile_dim0 = 128
        g1[4] = (int)128;           // tile_dim1 = 128
        g1[5] = (int)THREE_D;       // tensor_dim0_stride = 768
        g1[6] = 0;
        g1[7] = 0;

        i32x4 z4 = {0, 0, 0, 0};
#if __clang_major__ >= 23
        i32x8 z8 = {0, 0, 0, 0, 0, 0, 0, 0};
        __builtin_amdgcn_tensor_load_to_lds(g0, g1, z4, z4, z8, 0);
#else
        __builtin_amdgcn_tensor_load_to_lds(g0, g1, z4, z4, 0);
#endif
    }
#else
    for (int idx = tid; idx < 128 * 128; idx += 256) {
        int r = idx >> 7, c = idx & 127;
        Kl[r * KLD + c] = (r < len)
            ? qkv[(long)(row0 + r) * THREE_D + D_MODEL + h * HDIM + c] : 0.f;
    }
#endif

    // ---- stage V transposed: Vt[d][k_row] (coalesced global reads), overlaps TDM
    for (int idx = tid; idx < 128 * 128; idx += 256) {
        int d = idx & 127, r = idx >> 7;
        float vv = (r < len)
            ? qkv[(long)(row0 + r) * THREE_D + 2 * D_MODEL + h * HDIM + d] : 0.f;
        Vt[d * VTL + r] = vv;
    }

#if __has_builtin(__builtin_amdgcn_tensor_load_to_lds) && __has_builtin(__builtin_amdgcn_s_wait_tensorcnt)
    if (tid < 32) __builtin_amdgcn_s_wait_tensorcnt(0);
#endif
    __syncthreads();

    const int wave = tid >> 5;
    const int lane = tid & 31;
    const int ml   = lane & 15;
    const int kh   = lane >> 4;

    // ---- scores = Q * K^T * (1/sqrt(HD)); wave w -> rows [w*16, w*16+16)
    {
        v8f acc[8] = {};
        int qrow = row0 + wave * 16 + ml;
        if (qrow >= LTOT) qrow = LTOT - 1;           // clamp (rows >= len unused)
        const float* qbase = qkv + (long)qrow * THREE_D + h * HDIM + 2 * kh;
        v2f a_c = *(const v2f*)(qbase);
        for (int k = 0; k < HDIM; k += 4) {
            v2f a_n = a_c;
            if (k + 4 < HDIM) a_n = *(const v2f*)(qbase + k + 4);  // prefetch
            v2f bf[8];
#pragma unroll
            for (int t = 0; t < 8; ++t)
                bf[t] = *(const v2f*)(Kl + (t * 16 + ml) * KLD + k + 2 * kh);
#pragma unroll
            for (int t = 0; t < 8; ++t) acc[t] = wmma_f32(a_c, bf[t], acc[t]);
            a_c = a_n;
        }
        const float scale = 0.08838834764831845f;    // 1/sqrt(128)
#pragma unroll
        for (int t = 0; t < 8; ++t)
#pragma unroll
            for (int r = 0; r < 8; ++r) {
                int lr = wave * 16 + r + 8 * kh;
                S[lr * SLD + t * 16 + ml] = acc[t][r] * scale;
            }
    }
    __syncthreads();

    // ---- row softmax: 2 threads per row, 64 cols each; pair combine by shfl
    {
        const int r    = tid >> 1;
        const int half = tid & 1;
        float* srow = S + r * SLD + half * 64;
        float mx = -INFINITY;
        for (int n = 0; n < 64; n += 2) {
            v2f p = *(const v2f*)(srow + n);
            int c0 = half * 64 + n;
            if (c0 < len)     mx = fmaxf(mx, p.x);
            if (c0 + 1 < len) mx = fmaxf(mx, p.y);
        }
        mx = fmaxf(mx, __shfl_xor(mx, 1));           // lane pair: same wave
        float sum = 0.f;
        for (int n = 0; n < 64; ++n) {
            int c = half * 64 + n;
            float e = (c < len && r < len) ? __expf(srow[n] - mx) : 0.f;
            srow[n] = e; sum += e;
        }
        sum += __shfl_xor(sum, 1);
        float inv = (r < len) ? 1.f / sum : 0.f;
        for (int n = 0; n < 64; ++n) srow[n] *= inv;
    }
    __syncthreads();

    // ---- O = softmax(S) * V  (B fragments from transposed V: contiguous b64)
    {
        v8f acc[8] = {};
        const float* sbase = S + (wave * 16 + ml) * SLD + 2 * kh;
        for (int k = 0; k < 128; k += 4) {
            v2f a = *(const v2f*)(sbase + k);
            v2f bf[8];
#pragma unroll
            for (int t = 0; t < 8; ++t)
                bf[t] = *(const v2f*)(Vt + (t * 16 + ml) * VTL + k + 2 * kh);
#pragma unroll
            for (int t = 0; t < 8; ++t) acc[t] = wmma_f32(a, bf[t], acc[t]);
        }
#pragma unroll
        for (int t = 0; t < 8; ++t)
#pragma unroll
            for (int r = 0; r < 8; ++r) {
                int lr = wave * 16 + r + 8 * kh;
                if (lr < len)
                    O[(long)(row0 + lr) * D_MODEL + h * HDIM + t * 16 + ml] = acc[t][r];
            }
    }
}

// ---------------------------------------------------------------------------
// Fused: out = LayerNorm(xres + A*W^T + bias) * g + b
// grid = M/64 blocks; block = 256 (8 waves as 2x4). Each block owns a full
// 64x256 output tile (N=256 == D), so rows can be LayerNormed in-kernel.
// ---------------------------------------------------------------------------
template <int K>
__global__ __launch_bounds__(256) void gemm_out_ln_kernel(
    const float* __restrict__ A,      // [M, K] attention output
    const float* __restrict__ W,      // [256, K]
    const float* __restrict__ bias,   // [256]
    const float* __restrict__ xres,   // [M, 256] residual input
    const float* __restrict__ g, const float* __restrict__ bb,
    float* __restrict__ out)          // [M, 256]
{
    extern __shared__ float T[];      // [64][TLD]
    const int tid  = threadIdx.x;
    const int wave = tid >> 5;
    const int lane = tid & 31;
    const int ml   = lane & 15;
    const int kh   = lane >> 4;
    const int wm   = wave & 1, wn = wave >> 1;    // 2 row strips x 4 col strips
    const int m0   = blockIdx.x * 64;
    const int mw   = wm * 32;
    const int nw   = wn * 64;

    const float* a0 = A + (long)(m0 + mw + ml) * K + 2 * kh;
    const float* a1 = a0 + 16 * K;
    const float* w0 = W + (long)(nw + ml) * K + 2 * kh;

    v8f acc[2][4] = {};
    v2f ac0 = *(const v2f*)(a0);
    v2f ac1 = *(const v2f*)(a1);
    v2f bc[4];
#pragma unroll
    for (int t = 0; t < 4; ++t) bc[t] = *(const v2f*)(w0 + t * 16 * K);

    for (int k = 4; k < K; k += 4) {
        v2f an0 = *(const v2f*)(a0 + k);
        v2f an1 = *(const v2f*)(a1 + k);
        v2f bn[4];
#pragma unroll
        for (int t = 0; t < 4; ++t) bn[t] = *(const v2f*)(w0 + t * 16 * K + k);
#pragma unroll
        for (int t = 0; t < 4; ++t) {
            acc[0][t] = wmma_f32(ac0, bc[t], acc[0][t]);
            acc[1][t] = wmma_f32(ac1, bc[t], acc[1][t]);
        }
        ac0 = an0; ac1 = an1;
#pragma unroll
        for (int t = 0; t < 4; ++t) bc[t] = bn[t];
    }
#pragma unroll
    for (int t = 0; t < 4; ++t) {
        acc[0][t] = wmma_f32(ac0, bc[t], acc[0][t]);
        acc[1][t] = wmma_f32(ac1, bc[t], acc[1][t]);
    }

    // GEMM tile -> LDS (with bias)
#pragma unroll
    for (int t = 0; t < 4; ++t) {
        float bv = bias[nw + t * 16 + ml];
#pragma unroll
        for (int rm = 0; rm < 2; ++rm)
#pragma unroll
            for (int r = 0; r < 8; ++r)
                T[(mw + rm * 16 + r + 8 * kh) * TLD + nw + t * 16 + ml]
                    = acc[rm][t][r] + bv;
    }
    __syncthreads();

    // residual add (coalesced global reads)
    for (int idx = tid; idx < 64 * 256; idx += 256) {
        int r = idx >> 8, c = idx & 255;
        T[r * TLD + c] += xres[(long)(m0 + r) * D_MODEL + c];
    }
    __syncthreads();

    // LayerNorm: 4 threads per row; interleaved cols c = q + 4n so the four
    // lanes of a row hit distinct LDS banks; combines via shfl (same wave).
    {
        const int r = tid >> 2;
        const int q = tid & 3;
        const float* trow = T + r * TLD + q;
        float s = 0.f;
        for (int n = 0; n < 64; ++n) s += trow[4 * n];
        s += __shfl_xor(s, 1);
        s += __shfl_xor(s, 2);
        float mu = s * (1.0f / D_MODEL);
        float vs = 0.f;
        for (int n = 0; n < 64; ++n) { float d = trow[4 * n] - mu; vs += d * d; }
        vs += __shfl_xor(vs, 1);
        vs += __shfl_xor(vs, 2);
        float inv = rsqrtf(vs * (1.0f / D_MODEL) + EPSV);
        for (int n = 0; n < 64; ++n) {
            int c = q + 4 * n;
            out[(long)(m0 + r) * D_MODEL + c] = (trow[4 * n] - mu) * inv * g[c] + bb[c];
        }
    }
}

// ---------------------------------------------------------------------------
extern "C" void kernel_launch(void* const* d_in, const int* in_sizes, int n_in,
                              void* d_out, int out_size, void* d_ws, size_t ws_size,
                              hipStream_t stream) {
    (void)in_sizes; (void)n_in; (void)out_size; (void)ws_size;

    const float*         x0    = (const float*)d_in[0];
    const unsigned char* mask  = (const unsigned char*)d_in[1]; // jnp bool = 1 byte
    const float*         in_w  = (const float*)d_in[2];  // [2, 768, 256]
    const float*         in_b  = (const float*)d_in[3];  // [2, 768]
    const float*         out_w = (const float*)d_in[4];  // [2, 256, 256]
    const float*         out_b = (const float*)d_in[5];  // [2, 256]
    const float*         ln_g  = (const float*)d_in[6];  // [2, 256]
    const float*         ln_b  = (const float*)d_in[7];  // [2, 256]
    float* out = (float*)d_out;

    // workspace layout
    int*   ivals = (int*)d_ws;                       // starts[64], counts[64]
    float* fbase = (float*)d_ws + 256;
    float* qkv   = fbase;                            // [8192, 768]
    float* Obuf  = qkv + (long)LTOT * THREE_D;       // [8192, 256]

    prep_kernel<<<1, 64, 0, stream>>>(mask, ivals, ivals + 64);

    const float* xsrc = x0;
    for (int layer = 0; layer < 2; ++layer) {
        gemm_nt_bias<D_MODEL><<<dim3(LTOT / 128, THREE_D / 128), 256, 0, stream>>>(
            xsrc, D_MODEL,
            in_w + (long)layer * THREE_D * D_MODEL, in_b + (long)layer * THREE_D,
            qkv, THREE_D);

        attn_kernel<<<dim3(BSAMP, NHEAD), 256, ATTN_LDS_BYTES, stream>>>(
            qkv, Obuf, ivals, ivals + 64);

        gemm_out_ln_kernel<D_MODEL><<<LTOT / 64, 256, OUTLN_LDS_BYTES, stream>>>(
            Obuf,
            out_w + (long)layer * D_MODEL * D_MODEL, out_b + (long)layer * D_MODEL,
            xsrc,
            ln_g + (long)layer * D_MODEL, ln_b + (long)layer * D_MODEL,
            out);

        xsrc = out;
    }
}